// SinkAttentionRotaryImpl_12146167513324
// MI455X (gfx1250) — compile-verified
//
#include <hip/hip_runtime.h>
#include <hip/hip_bf16.h>
#include <stdint.h>
#include <stddef.h>

// ---------------------------------------------------------------------------
// Problem constants (from the reference)
// ---------------------------------------------------------------------------
#define KC_FLOATS     67108864ull        // 4096 blocks * 8 heads * 16*16*8
#define KC_CHUNK16    16777216ull        // KC_FLOATS / 4  (16-byte chunks)
#define BLK_FLOATS    2048               // per (block, head): 16*16*8 floats
#define NUM_TABLES    257
#define CACHE_SIZE_F  4112.0f            // SLIDING_WINDOW + SINK_SIZE

// ---------------------------------------------------------------------------
// gfx1250 async global<->LDS copy path (ASYNCcnt-tracked DMA-style copy)
// ---------------------------------------------------------------------------
#if defined(__HIP_DEVICE_COMPILE__) && defined(__gfx1250__)
  #if __has_builtin(__builtin_amdgcn_global_load_async_to_lds_b128) && \
      __has_builtin(__builtin_amdgcn_global_store_async_from_lds_b128)
    #define USE_ASYNC_LDS_COPY 1
  #endif
#endif

#if defined(USE_ASYNC_LDS_COPY)
typedef int v4i __attribute__((ext_vector_type(4)));
typedef __attribute__((address_space(1))) v4i GV4I;   // global 128-bit chunk
typedef __attribute__((address_space(3))) v4i LV4I;   // LDS    128-bit chunk
#endif

__device__ __forceinline__ void wait_async_zero() {
#if defined(__gfx1250__)
  #if __has_builtin(__builtin_amdgcn_s_wait_asynccnt)
    __builtin_amdgcn_s_wait_asynccnt(0);
  #else
    asm volatile("s_wait_asynccnt 0x0" ::: "memory");
  #endif
#endif
}

#define CP_BLOCK 256                     // threads per workgroup (8 waves)
#define CP_U     8                       // 16B chunks per thread (32 KB / WG)
#define CP_GRID  8192                    // KC_CHUNK16 / (CP_BLOCK * CP_U)

// Single-shot bulk copy: each WG moves exactly 32 KB through its private LDS
// staging buffer. Every LDS byte is written once (async load) and read once
// (async store) by the same lane, separated by s_wait_asynccnt 0 -> no WAR
// hazard on the staging buffer, no barriers needed.
__global__ void __launch_bounds__(CP_BLOCK)
sink_bulk_copy_kernel(const float* __restrict__ src,
                      float* __restrict__ dst,
                      unsigned long long nchunks) {
#if defined(USE_ASYNC_LDS_COPY)
  __shared__ alignas(16) float lds_buf[CP_BLOCK * CP_U * 4];   // 32 KB staging
  const unsigned long long base =
      (unsigned long long)blockIdx.x * (CP_BLOCK * CP_U);
#pragma unroll
  for (int u = 0; u < CP_U; ++u) {
    const unsigned long long c = base + (unsigned long long)u * CP_BLOCK + threadIdx.x;
    __builtin_amdgcn_global_load_async_to_lds_b128(
        (GV4I*)(src + 4ull * c),
        (LV4I*)&lds_buf[(u * CP_BLOCK + threadIdx.x) * 4],
        /*offset=*/0, /*cpol=*/0);
  }
  // All 8 loads of this wave landed in LDS (per-wave ASYNCcnt).
  wait_async_zero();
#pragma unroll
  for (int u = 0; u < CP_U; ++u) {
    const unsigned long long c = base + (unsigned long long)u * CP_BLOCK + threadIdx.x;
    __builtin_amdgcn_global_store_async_from_lds_b128(
        (GV4I*)(dst + 4ull * c),
        (LV4I*)&lds_buf[(u * CP_BLOCK + threadIdx.x) * 4],
        /*offset=*/0, /*cpol=*/0);
  }
  // s_endpgm performs an implicit wait-idle, draining the async stores.
  (void)nchunks;
#else
  // Fallback: plain float4 grid-stride copy (still saturates HBM).
  const float4* __restrict__ s4 = (const float4*)src;
  float4* __restrict__ d4 = (float4*)dst;
  unsigned long long i = (unsigned long long)blockIdx.x * blockDim.x + threadIdx.x;
  const unsigned long long stride = (unsigned long long)gridDim.x * blockDim.x;
  for (; i < nchunks; i += stride) d4[i] = s4[i];
#endif
}

// ---------------------------------------------------------------------------
// Sink-block RoPE rotate + backup + scatter-overwrite.
// One workgroup per (batch, head); 64*8 = 512 workgroups.
// ---------------------------------------------------------------------------
__global__ void __launch_bounds__(256)
sink_rope_kernel(const float* __restrict__ key_cache,
                 const int* __restrict__ block_tables,
                 const int* __restrict__ positions,
                 float* __restrict__ out) {
  const int bh = blockIdx.x;           // 0..511
  const int b = bh >> 3;               // batch 0..63
  const int h = bh & 7;                // head  0..7

  const int blk = block_tables[b * NUM_TABLES];  // sink block (column 0)
  // num_evictions = int(max(pos - CACHE_SIZE, 0)) used as float
  float ne = fmaxf((float)positions[b] - CACHE_SIZE_F, 0.0f);
  ne = (float)(int)ne;

  const size_t cache_off = ((size_t)blk * 8 + h) * BLK_FLOATS;
  const float* __restrict__ src  = key_cache + cache_off;
  float* __restrict__ dst_cache  = out + cache_off;                       // rotated scatter
  float* __restrict__ dst_backup = out + KC_FLOATS
                                 + ((size_t)b * 8 + h) * BLK_FLOATS;      // raw backup

  // 1024 (dim, token) pairs: d in [0,64), t in [0,16). Element layout inside a
  // (block, head) tile is [HEAD_SIZE/8=16][BLOCK=16][8], so
  // idx(d,t) = (d/8)*128 + t*8 + (d%8); the paired dim d+64 sits at idx+1024.
  for (int p = threadIdx.x; p < 1024; p += blockDim.x) {
    const int d = p >> 4;
    const int t = p & 15;
    const int idx = ((d >> 3) << 7) + (t << 3) + (d & 7);

    const float x1 = src[idx];
    const float x2 = src[idx + 1024];

    dst_backup[idx]        = x1;
    dst_backup[idx + 1024] = x2;

    // inv_freq[d] = 10000^(-d/64) = exp(-d * ln(10000)/64)
    const float invf = __expf(-(float)d * 0.14391156831212787f);
    float s, c;
    sincosf(ne * invf, &s, &c);

    dst_cache[idx]        = x1 * c - x2 * s;
    dst_cache[idx + 1024] = x2 * c + x1 * s;
  }
}

// ---------------------------------------------------------------------------
// Launch
// ---------------------------------------------------------------------------
extern "C" void kernel_launch(void* const* d_in, const int* in_sizes, int n_in,
                              void* d_out, int out_size, void* d_ws, size_t ws_size,
                              hipStream_t stream) {
  const float* key_cache    = (const float*)d_in[0];
  const int*   block_tables = (const int*)d_in[1];
  const int*   positions    = (const int*)d_in[2];
  float*       out          = (float*)d_out;

  // 1) Bulk copy key_cache -> new_key_cache region of d_out (~256 MB each way;
  //    this is the entire cost of the op: ~516 MB / 23.3 TB/s ~= 22 us).
  //    CP_GRID * CP_BLOCK * CP_U 16B-chunks == KC_CHUNK16 exactly.
  sink_bulk_copy_kernel<<<CP_GRID, CP_BLOCK, 0, stream>>>(key_cache, out, KC_CHUNK16);

  // 2) RoPE-rotate the 64 sink blocks in place in d_out and emit the backup.
  //    Stream ordering guarantees this lands after the bulk copy.
  sink_rope_kernel<<<512, 256, 0, stream>>>(key_cache, block_tables, positions, out);

  (void)in_sizes; (void)n_in; (void)out_size; (void)d_ws; (void)ws_size;
}